// VolumeRenderer_53618371723799
// MI455X (gfx1250) — compile-verified
//
#include <hip/hip_runtime.h>
#include <hip/hip_bf16.h>

typedef float v2f __attribute__((ext_vector_type(2)));
typedef float v8f __attribute__((ext_vector_type(8)));

// ---------------- threefry2x32 (matches jax.random, key = [0,1]) -------------
__device__ __forceinline__ unsigned rotl32(unsigned x, int r) {
  return (x << r) | (x >> (32 - r));
}

__device__ __forceinline__ void threefry2x32(unsigned k0, unsigned k1,
                                             unsigned& x0, unsigned& x1) {
  const unsigned ks2 = k0 ^ k1 ^ 0x1BD11BDAu;
  const int ra[4] = {13, 15, 26, 6};
  const int rb[4] = {17, 29, 16, 24};
  x0 += k0; x1 += k1;
#pragma unroll
  for (int i = 0; i < 4; ++i) { x0 += x1; x1 = rotl32(x1, ra[i]); x1 ^= x0; }
  x0 += k1; x1 += ks2 + 1u;
#pragma unroll
  for (int i = 0; i < 4; ++i) { x0 += x1; x1 = rotl32(x1, rb[i]); x1 ^= x0; }
  x0 += ks2; x1 += k0 + 2u;
#pragma unroll
  for (int i = 0; i < 4; ++i) { x0 += x1; x1 = rotl32(x1, ra[i]); x1 ^= x0; }
  x0 += k0; x1 += k1 + 3u;
#pragma unroll
  for (int i = 0; i < 4; ++i) { x0 += x1; x1 = rotl32(x1, rb[i]); x1 ^= x0; }
  x0 += k1; x1 += ks2 + 4u;
#pragma unroll
  for (int i = 0; i < 4; ++i) { x0 += x1; x1 = rotl32(x1, ra[i]); x1 ^= x0; }
  x0 += ks2; x1 += k0 + 5u;
}

// ---- pos-enc GEMM for groups of 64 samples: E[64x64] staged in LDS, then
//      each wave does one 16-sample tile with V_WMMA_F32_16X16X4_F32.
// A (16x4 f32): lane L -> row m=L&15, K = khalf*2+{0,1} per chunk.
// B (4x16 f32): lane L -> col n=L&15, same K halving. C/D: M = j + 8*khalf.
__device__ void run_net(const float* o, const float* dv,
                        const float* z, int nS,
                        const float* Wt,   // [5][64] LDS (row 4 = zeros)
                        const float* dc,   // [4]  LDS: bias + dir contribution
                        float* optic,      // [nS][4] LDS
                        float* emb,        // [64][64] LDS staging
                        int tid) {
  const int lane = tid & 31;
  const int wave = tid >> 5;
  const int m = lane & 15;
  const int khalf = lane >> 4;
  const int n = m;
  const int wrow = (n < 4) ? n : 4;
  const float* brow = Wt + wrow * 64 + khalf * 2;
  const int ngrp = nS >> 6;

  for (int grp = 0; grp < ngrp; ++grp) {
    const int sbase = grp << 6;
    // --- stage raw coords + zero pad (k = 0..2, 63), 256 entries ---
    for (int idx = tid; idx < 256; idx += 128) {
      int s = idx >> 2, c = idx & 3;
      float zv = z[sbase + s];
      emb[s * 64 + ((c < 3) ? c : 63)] = (c < 3) ? fmaf(dv[c], zv, o[c]) : 0.0f;
    }
    // --- stage sin/cos pairs: 64 samples x 10 freqs x 3 coords ---
    for (int idx = tid; idx < 1920; idx += 128) {
      int s = idx / 30, kk = idx - 30 * s;
      int f = kk / 3, c = kk - 3 * f;
      float zv = z[sbase + s];
      float pv = fmaf(dv[c], zv, o[c]);
      float sv, cv;
      sincosf(ldexpf(pv, f), &sv, &cv);
      emb[s * 64 + 3 + 6 * f + c] = sv;   // sin block of this freq
      emb[s * 64 + 6 + 6 * f + c] = cv;   // cos block of this freq
    }
    __syncthreads();

    // --- WMMA: wave w handles samples [sbase + w*16, +16) ---
    const float* arow = emb + (wave * 16 + m) * 64 + khalf * 2;
    v8f acc = {0.f, 0.f, 0.f, 0.f, 0.f, 0.f, 0.f, 0.f};
#pragma unroll
    for (int chunk = 0; chunk < 16; ++chunk) {
      v2f a  = *(const v2f*)(arow + chunk * 4);
      v2f bb = *(const v2f*)(brow + chunk * 4);
      acc = __builtin_amdgcn_wmma_f32_16x16x4_f32(
          false, a, false, bb, (short)0, acc, false, false);
    }
    if (n < 4) {
#pragma unroll
      for (int j = 0; j < 8; ++j)
        optic[(sbase + wave * 16 + khalf * 8 + j) * 4 + n] = acc[j] + dc[n];
    }
    __syncthreads();
  }
}

// ---------------------------- megakernel: 1 block = 1 ray --------------------
__global__ __launch_bounds__(128) void VolumeRenderer_kernel(
    const float* __restrict__ rays, const float* __restrict__ Wc,
    const float* __restrict__ bc, const float* __restrict__ Wf,
    const float* __restrict__ bf, float* __restrict__ out, int M) {
  __shared__ float sh_emb[64 * 64];     // 16 KB staging
  __shared__ float sh_optic[192 * 4];
  __shared__ float sh_Wt[5 * 64];       // transposed pos-part weights (+zero row)
  __shared__ float sh_z[64];
  __shared__ float sh_w[64];
  __shared__ float sh_zmid[63];
  __shared__ float sh_cdf[63];
  __shared__ float sh_allz[192];
  __shared__ float sh_sorted[192];
  __shared__ float sh_dir[27];
  __shared__ float sh_dc[4];

  const int ray = blockIdx.x;
  const int tid = threadIdx.x;

  float o[3], dv[3];
  o[0] = rays[ray * 3 + 0]; o[1] = rays[ray * 3 + 1]; o[2] = rays[ray * 3 + 2];
  const float* rd = rays + (size_t)M * 3;
  dv[0] = rd[ray * 3 + 0]; dv[1] = rd[ray * 3 + 1]; dv[2] = rd[ray * 3 + 2];
  float dnorm = sqrtf(dv[0] * dv[0] + dv[1] * dv[1] + dv[2] * dv[2]);
  float vd[3] = {dv[0] / dnorm, dv[1] / dnorm, dv[2] / dnorm};

  // ---- per-ray setup: Wt(coarse), dir encoding, coarse z ----
  for (int i = tid; i < 320; i += 128) {
    int r = i >> 6, k = i & 63;
    sh_Wt[i] = (r < 4 && k < 63) ? Wc[k * 4 + r] : 0.0f;
  }
  if (tid < 12) {
    int f = tid / 3, c = tid - 3 * f;
    float sv, cv;
    sincosf(ldexpf(vd[c], f), &sv, &cv);
    sh_dir[3 + 6 * f + c] = sv;
    sh_dir[6 + 6 * f + c] = cv;
  } else if (tid < 15) {
    sh_dir[tid - 12] = vd[tid - 12];
  }
  if (tid < 64) sh_z[tid] = 2.0f + 4.0f * (float)tid / 63.0f;  // NEAR..FAR
  __syncthreads();

  // dir-part contribution folded with bias: dc[n] = bc[n] + dir_emb . Wc[63:90,n]
  if (tid < 4) {
    float a = bc[tid];
    for (int k = 0; k < 27; ++k) a += sh_dir[k] * Wc[(63 + k) * 4 + tid];
    sh_dc[tid] = a;
  }
  // (visible after run_net's internal first __syncthreads)

  // ---- coarse network ----
  run_net(o, dv, sh_z, 64, sh_Wt, sh_dc, sh_optic, sh_emb, tid);

  // ---- coarse composite phase 1 (parallel transcendentals) ----
  for (int i = tid; i < 64; i += 128) {
    float dist = ((i < 63) ? (sh_z[i + 1] - sh_z[i]) : 1e10f) * dnorm;
    float sigma = fmaxf(sh_optic[i * 4 + 3], 0.0f);
    float alpha = 1.0f - expf(-sigma * dist);
    sh_optic[i * 4 + 0] = 1.0f / (1.0f + expf(-sh_optic[i * 4 + 0]));
    sh_optic[i * 4 + 1] = 1.0f / (1.0f + expf(-sh_optic[i * 4 + 1]));
    sh_optic[i * 4 + 2] = 1.0f / (1.0f + expf(-sh_optic[i * 4 + 2]));
    sh_optic[i * 4 + 3] = alpha;
  }
  if (tid < 63) sh_zmid[tid] = 0.5f * (sh_z[tid] + sh_z[tid + 1]);
  __syncthreads();

  // ---- coarse scan (cumprod) + cdf, serial on thread 0 ----
  if (tid == 0) {
    float T = 1.0f, r0 = 0.f, r1 = 0.f, r2 = 0.f, wsum = 0.f;
    for (int i = 0; i < 64; ++i) {
      float alpha = sh_optic[i * 4 + 3];
      float w = alpha * T;
      sh_w[i] = w;
      r0 += w * sh_optic[i * 4 + 0];
      r1 += w * sh_optic[i * 4 + 1];
      r2 += w * sh_optic[i * 4 + 2];
      wsum += w;
      T *= (1.0f - alpha + 1e-10f);
    }
    float bg = 1.0f - wsum;  // WHITE_BKGD
    size_t co = (size_t)M * 3 + (size_t)ray * 3;
    out[co + 0] = r0 + bg; out[co + 1] = r1 + bg; out[co + 2] = r2 + bg;
    float sum = 0.0f;
    for (int i = 0; i < 62; ++i) sum += sh_w[i + 1] + 1e-5f;
    float run = 0.0f;
    sh_cdf[0] = 0.0f;
    for (int i = 0; i < 62; ++i) {
      run += (sh_w[i + 1] + 1e-5f) / sum;
      sh_cdf[i + 1] = run;
    }
  }
  __syncthreads();

  // ---- importance sampling: exact jax.random.uniform(key(1),(M,128)) ----
  {
    unsigned jf = (unsigned)ray * 128u + (unsigned)tid;
    unsigned half = (unsigned)M * 64u;          // (M*128)/2
    unsigned x0 = (jf < half) ? jf : (jf - half);
    unsigned x1 = (jf < half) ? (jf + half) : jf;
    threefry2x32(0u, 1u, x0, x1);
    unsigned bits = (jf < half) ? x0 : x1;
    float u = __uint_as_float((bits >> 9) | 0x3f800000u) - 1.0f;
    int inds = 0;
    for (int i = 0; i < 63; ++i) inds += (sh_cdf[i] <= u) ? 1 : 0;  // side='right'
    int below = inds - 1; if (below < 0) below = 0;
    int above = inds;     if (above > 62) above = 62;
    float cb = sh_cdf[below], ca = sh_cdf[above];
    float zb = sh_zmid[below], za = sh_zmid[above];
    float denom = ca - cb;
    if (denom < 1e-5f) denom = 1.0f;
    float tt = (u - cb) / denom;
    sh_allz[64 + tid] = zb + tt * (za - zb);
  }
  if (tid < 64) sh_allz[tid] = sh_z[tid];
  // swap in fine weights + fine dir contribution
  for (int i = tid; i < 320; i += 128) {
    int r = i >> 6, k = i & 63;
    sh_Wt[i] = (r < 4 && k < 63) ? Wf[k * 4 + r] : 0.0f;
  }
  if (tid < 4) {
    float a = bf[tid];
    for (int k = 0; k < 27; ++k) a += sh_dir[k] * Wf[(63 + k) * 4 + tid];
    sh_dc[tid] = a;
  }
  __syncthreads();

  // ---- sort 192 z values by rank counting ----
  for (int i = tid; i < 192; i += 128) {
    float v = sh_allz[i];
    int rank = 0;
    for (int j = 0; j < 192; ++j) {
      float vj = sh_allz[j];
      rank += (vj < v || (vj == v && j < i)) ? 1 : 0;
    }
    sh_sorted[rank] = v;
  }
  __syncthreads();

  // ---- fine network ----
  run_net(o, dv, sh_sorted, 192, sh_Wt, sh_dc, sh_optic, sh_emb, tid);

  // ---- fine composite ----
  for (int i = tid; i < 192; i += 128) {
    float dist = ((i < 191) ? (sh_sorted[i + 1] - sh_sorted[i]) : 1e10f) * dnorm;
    float sigma = fmaxf(sh_optic[i * 4 + 3], 0.0f);
    float alpha = 1.0f - expf(-sigma * dist);
    sh_optic[i * 4 + 0] = 1.0f / (1.0f + expf(-sh_optic[i * 4 + 0]));
    sh_optic[i * 4 + 1] = 1.0f / (1.0f + expf(-sh_optic[i * 4 + 1]));
    sh_optic[i * 4 + 2] = 1.0f / (1.0f + expf(-sh_optic[i * 4 + 2]));
    sh_optic[i * 4 + 3] = alpha;
  }
  __syncthreads();
  if (tid == 0) {
    float T = 1.0f, r0 = 0.f, r1 = 0.f, r2 = 0.f, wsum = 0.f;
    for (int i = 0; i < 192; ++i) {
      float alpha = sh_optic[i * 4 + 3];
      float w = alpha * T;
      r0 += w * sh_optic[i * 4 + 0];
      r1 += w * sh_optic[i * 4 + 1];
      r2 += w * sh_optic[i * 4 + 2];
      wsum += w;
      T *= (1.0f - alpha + 1e-10f);
    }
    float bg = 1.0f - wsum;
    size_t fo = (size_t)ray * 3;
    out[fo + 0] = r0 + bg; out[fo + 1] = r1 + bg; out[fo + 2] = r2 + bg;
  }
}

extern "C" void kernel_launch(void* const* d_in, const int* in_sizes, int n_in,
                              void* d_out, int out_size, void* d_ws, size_t ws_size,
                              hipStream_t stream) {
  const float* rays = (const float*)d_in[0];
  const float* Wc   = (const float*)d_in[1];
  const float* bc   = (const float*)d_in[2];
  const float* Wf   = (const float*)d_in[3];
  const float* bf   = (const float*)d_in[4];
  float* out = (float*)d_out;
  int M = in_sizes[0] / 6;  // rays is [2, M, 3]
  hipLaunchKernelGGL(VolumeRenderer_kernel, dim3(M), dim3(128), 0, stream,
                     rays, Wc, bc, Wf, bf, out, M);
}